// GNN_16338055594318
// MI455X (gfx1250) — compile-verified
//
#include <hip/hip_runtime.h>
#include <hip/hip_bf16.h>

typedef float v2f __attribute__((ext_vector_type(2)));
typedef float v8f __attribute__((ext_vector_type(8)));
typedef unsigned int u32x4 __attribute__((ext_vector_type(4)));
typedef int i32x4 __attribute__((ext_vector_type(4)));
typedef int i32x8 __attribute__((ext_vector_type(8)));

#define DD 128          // embedding dim
#define DD2 256         // 2*D
#define NLAYER 5
#define BN_EPS 1e-5f

// ---------------------------------------------------------------------------
// h[n, :] = x_emb1[x[n,0], :] + x_emb2[x[n,1], :]     (float4-vectorized)
// ---------------------------------------------------------------------------
__global__ __launch_bounds__(256)
void embed_kernel(const int* __restrict__ x,
                  const float* __restrict__ e1,
                  const float* __restrict__ e2,
                  float* __restrict__ h, int N)
{
    int i = blockIdx.x * blockDim.x + threadIdx.x;          // over N * (D/4)
    if (i >= N * (DD / 4)) return;
    int n  = i >> 5;                                        // D/4 == 32
    int c4 = i & 31;
    int a = x[2 * n], b = x[2 * n + 1];
    float4 v1 = ((const float4*)(e1 + (size_t)a * DD))[c4];
    float4 v2 = ((const float4*)(e2 + (size_t)b * DD))[c4];
    float4 o;
    o.x = v1.x + v2.x; o.y = v1.y + v2.y;
    o.z = v1.z + v2.z; o.w = v1.w + v2.w;
    ((float4*)(h + (size_t)n * DD))[c4] = o;
}

// ---------------------------------------------------------------------------
// zero fill (float4)
// ---------------------------------------------------------------------------
__global__ __launch_bounds__(256)
void zero_kernel(float4* __restrict__ p, size_t n4)
{
    size_t i = (size_t)blockIdx.x * blockDim.x + threadIdx.x;
    if (i < n4) p[i] = make_float4(0.f, 0.f, 0.f, 0.f);
}

// ---------------------------------------------------------------------------
// One wave32 per edge (E real edges + N self-loops).
// lane c handles channels [4c, 4c+4):  agg[dst] += h[src] + ee1[a0] + ee2[a1]
// ---------------------------------------------------------------------------
__global__ __launch_bounds__(256)
void scatter_kernel(const int* __restrict__ edge_index,
                    const int* __restrict__ edge_attr,
                    const float* __restrict__ ee1,   // [6, D] this layer
                    const float* __restrict__ ee2,   // [3, D] this layer
                    const float* __restrict__ h,
                    float* __restrict__ agg,
                    int E, int N)
{
    long long gid  = (long long)blockIdx.x * blockDim.x + threadIdx.x;
    int edge = (int)(gid >> 5);
    int lane = (int)(gid & 31);
    int total = E + N;
    if (edge >= total) return;

    int s, dvtx, a0, a1;
    if (edge < E) {
        if (lane == 0 && edge + 1024 < E)
            __builtin_prefetch(edge_index + edge + 1024, 0, 0);   // global_prefetch_b8
        s    = edge_index[edge];
        dvtx = edge_index[E + edge];
        a0   = edge_attr[2 * edge];
        a1   = edge_attr[2 * edge + 1];
    } else {                      // self-loop: attr = [4, 0]
        s = dvtx = edge - E;
        a0 = 4; a1 = 0;
    }

    int c = lane * 4;
    float4 hv = *(const float4*)(h   + (size_t)s  * DD + c);
    float4 v1 = *(const float4*)(ee1 + (size_t)a0 * DD + c);
    float4 v2 = *(const float4*)(ee2 + (size_t)a1 * DD + c);
    float* dst = agg + (size_t)dvtx * DD + c;
    atomicAdd(dst + 0, hv.x + v1.x + v2.x);
    atomicAdd(dst + 1, hv.y + v1.y + v2.y);
    atomicAdd(dst + 2, hv.z + v1.z + v2.z);
    atomicAdd(dst + 3, hv.w + v1.w + v2.w);
}

// ---------------------------------------------------------------------------
// C[M,NCOLS] = (relu?)(A[M,K] @ B[K,NCOLS] + bias)  via V_WMMA_F32_16X16X4_F32
//
// Block = 128 threads = 4 waves.  Block owns a 16 (M) x 64 (N) output tile:
//   mtile = blockIdx.x (M = 50000 = 3125*16, exact), wave w = local n-tile.
// The K x 64 panel of B is DMA'd into LDS once per block by the Tensor Data
// Mover (TENSOR_LOAD_TO_LDS, tracked by TENSORcnt), then all 4 waves read
// their B fragments from LDS (ds_load) while A fragments stream from global
// (L2-resident).  EXEC is all-ones wherever WMMA executes.
//
// fp32 A-fragment layout (ISA 7.12.2, 32-bit A 16x4): lanes 0-15 hold
// {K=k, k+1} for row M=lane; lanes 16-31 hold {K=k+2, k+3} -> contiguous
// float2 load per lane.  B (4x16) is row-striped across lanes per VGPR.
// C/D: VGPR v, lane l -> row = v + (l>=16 ? 8 : 0), col = l%16.
// ---------------------------------------------------------------------------
template<int K, int NCOLS, bool RELU>
__global__ __launch_bounds__(128)
void gemm_tdm_kernel(const float* __restrict__ A,
                     const float* __restrict__ B,
                     const float* __restrict__ bias,
                     float* __restrict__ C)
{
    __shared__ float sB[K * 64];          // K=256 -> 64 KB (WGP has 320 KB)

    const int lane  = threadIdx.x & 31;
    const int wave  = threadIdx.x >> 5;                    // local n-tile 0..3
    const int mtile = blockIdx.x;
    const int koff  = (lane >> 4) * 2;                     // 0 or 2
    const int row_a = mtile * 16 + (lane & 15);
    const int coll  = wave * 16 + (lane & 15);             // column in LDS panel
    const int colg  = blockIdx.y * 64 + coll;              // global column

    // ---- TDM: stage B[K x 64] panel into LDS (issued by wave 0 only; the
    //      TDM ignores EXEC, so exactly one wave must own the descriptor) ----
    if (wave == 0) {
        unsigned long long gaddr =
            (unsigned long long)(size_t)(B + (size_t)blockIdx.y * 64);
        unsigned lds = (unsigned)(size_t)(&sB[0]);
        // D# group 0: count=1 user descriptor, lds_addr, 57-bit global addr,
        //             type=2 ("image") in bits [127:126]
        u32x4 g0 = { 1u,
                     lds,
                     (unsigned)(gaddr & 0xffffffffull),
                     (unsigned)((gaddr >> 32) & 0x1ffffffull) | (2u << 30) };
        // D# group 1: workgroup_mask=0, data_size=2 (4 bytes),
        //             tensor_dim0=NCOLS, tensor_dim1=K,
        //             tile_dim0=64, tile_dim1=K, tile_dim2=0,
        //             tensor_dim0_stride=NCOLS, tensor_dim1_stride=0
        i32x8 g1;
        g1[0] = (int)(2u << 16);
        g1[1] = (int)((unsigned)(NCOLS & 0xffff) << 16);
        g1[2] = (int)(((unsigned)NCOLS >> 16) | ((unsigned)(K & 0xffff) << 16));
        g1[3] = (int)(((unsigned)K >> 16) | (64u << 16));
        g1[4] = (int)(unsigned)(K & 0xffff);
        g1[5] = (int)(unsigned)NCOLS;
        g1[6] = 0;
        g1[7] = 0;
        i32x4 z4 = {0, 0, 0, 0};
#if __clang_major__ >= 23
        i32x8 z8 = {0, 0, 0, 0, 0, 0, 0, 0};
        __builtin_amdgcn_tensor_load_to_lds(g0, g1, z4, z4, z8, 0);
#else
        __builtin_amdgcn_tensor_load_to_lds(g0, g1, z4, z4, 0);
#endif
        __builtin_amdgcn_s_wait_tensorcnt(0);   // s_wait_tensorcnt 0
    }
    __syncthreads();

    const float* __restrict__ arow = A + (size_t)row_a * K + koff;
    const float* __restrict__ bcol = sB + coll + (size_t)koff * 64;

    v8f acc = {0.f, 0.f, 0.f, 0.f, 0.f, 0.f, 0.f, 0.f};

#pragma unroll
    for (int k = 0; k < K; k += 4) {
        v2f a, b;
        a.x = arow[k];                 // contiguous float2 from global (L2)
        a.y = arow[k + 1];
        b.x = bcol[k * 64];            // ds_load from the staged panel
        b.y = bcol[(k + 1) * 64];
        // D = A(16x4) x B(4x16) + C   -> v_wmma_f32_16x16x4_f32
        acc = __builtin_amdgcn_wmma_f32_16x16x4_f32(
                  /*neg_a=*/false, a, /*neg_b=*/false, b,
                  /*c_mod=*/(short)0, acc, /*reuse_a=*/false, /*reuse_b=*/false);
    }

    const float bv = bias[colg];
#pragma unroll
    for (int v = 0; v < 8; ++v) {
        int r = mtile * 16 + v + ((lane >> 4) << 3);
        float val = acc[v] + bv;
        if (RELU) val = fmaxf(val, 0.f);
        C[(size_t)r * NCOLS + colg] = val;
    }
}

// ---------------------------------------------------------------------------
// BatchNorm training stats: one block per channel, reduce over N rows.
// stats[c] = mean, stats[D + c] = rsqrt(var + eps)
// ---------------------------------------------------------------------------
__global__ __launch_bounds__(256)
void bn_stats_kernel(const float* __restrict__ hn,
                     float* __restrict__ stats, int N)
{
    const int c = blockIdx.x;
    const int t = threadIdx.x;
    float s = 0.f, sq = 0.f;
    for (int r = t; r < N; r += 256) {
        float v = hn[(size_t)r * DD + c];
        s += v; sq += v * v;
    }
    __shared__ float ss[256];
    __shared__ float ssq[256];
    ss[t] = s; ssq[t] = sq;
    __syncthreads();
    for (int off = 128; off > 0; off >>= 1) {
        if (t < off) { ss[t] += ss[t + off]; ssq[t] += ssq[t + off]; }
        __syncthreads();
    }
    if (t == 0) {
        float mean = ss[0] / (float)N;
        float var  = ssq[0] / (float)N - mean * mean;
        stats[c]      = mean;
        stats[DD + c] = rsqrtf(var + BN_EPS);
    }
}

// ---------------------------------------------------------------------------
// out = (hn - mean) * rstd * gamma + beta   (+ relu for l < L-1)
// ---------------------------------------------------------------------------
__global__ __launch_bounds__(256)
void bn_apply_kernel(const float* __restrict__ hn,
                     const float* __restrict__ stats,
                     const float* __restrict__ gamma,
                     const float* __restrict__ beta,
                     float* __restrict__ out, int N, int do_relu)
{
    int i = blockIdx.x * blockDim.x + threadIdx.x;      // over N * (D/4)
    if (i >= N * (DD / 4)) return;
    int c4 = i & 31;
    float4 v = ((const float4*)hn)[i];
    float4 m = ((const float4*)stats)[c4];
    float4 r = ((const float4*)(stats + DD))[c4];
    float4 g = ((const float4*)gamma)[c4];
    float4 b = ((const float4*)beta)[c4];
    float4 o;
    o.x = (v.x - m.x) * r.x * g.x + b.x;
    o.y = (v.y - m.y) * r.y * g.y + b.y;
    o.z = (v.z - m.z) * r.z * g.z + b.z;
    o.w = (v.w - m.w) * r.w * g.w + b.w;
    if (do_relu) {
        o.x = fmaxf(o.x, 0.f); o.y = fmaxf(o.y, 0.f);
        o.z = fmaxf(o.z, 0.f); o.w = fmaxf(o.w, 0.f);
    }
    ((float4*)out)[i] = o;
}

// ---------------------------------------------------------------------------
extern "C" void kernel_launch(void* const* d_in, const int* in_sizes, int n_in,
                              void* d_out, int out_size, void* d_ws, size_t ws_size,
                              hipStream_t stream)
{
    const int*   x         = (const int*)d_in[0];
    const int*   edge_index= (const int*)d_in[1];
    const int*   edge_attr = (const int*)d_in[2];
    const float* x_emb1    = (const float*)d_in[3];
    const float* x_emb2    = (const float*)d_in[4];
    const float* edge_emb1 = (const float*)d_in[5];   // [L,6,D]
    const float* edge_emb2 = (const float*)d_in[6];   // [L,3,D]
    const float* W1        = (const float*)d_in[7];   // [L,D,2D]
    const float* b1        = (const float*)d_in[8];   // [L,2D]
    const float* W2        = (const float*)d_in[9];   // [L,2D,D]
    const float* b2        = (const float*)d_in[10];  // [L,D]
    const float* gamma     = (const float*)d_in[11];  // [L,D]
    const float* beta      = (const float*)d_in[12];  // [L,D]

    const int N = in_sizes[0] / 2;
    const int E = in_sizes[1] / 2;

    float* ws    = (float*)d_ws;
    float* h     = ws;                          // N*D
    float* agg   = ws + (size_t)N * DD;         // N*D
    float* hm    = ws + 2 * (size_t)N * DD;     // N*2D
    float* hn    = ws + 4 * (size_t)N * DD;     // N*D
    float* stats = ws + 5 * (size_t)N * DD;     // 2*D

    // node input embedding
    {
        int tot = N * (DD / 4);
        embed_kernel<<<(tot + 255) / 256, 256, 0, stream>>>(x, x_emb1, x_emb2, h, N);
    }

    for (int l = 0; l < NLAYER; ++l) {
        // agg = 0
        size_t n4 = (size_t)N * (DD / 4);
        zero_kernel<<<(unsigned)((n4 + 255) / 256), 256, 0, stream>>>((float4*)agg, n4);

        // gather + message + scatter-add (edges + self-loops)
        long long th = (long long)(E + N) * 32;
        scatter_kernel<<<(unsigned)((th + 255) / 256), 256, 0, stream>>>(
            edge_index, edge_attr,
            edge_emb1 + (size_t)l * 6 * DD,
            edge_emb2 + (size_t)l * 3 * DD,
            h, agg, E, N);

        // hm = relu(agg @ W1[l] + b1[l])   [N,128] x [128,256]
        dim3 g1(N / 16, DD2 / 64);
        gemm_tdm_kernel<DD, DD2, true><<<g1, 128, 0, stream>>>(
            agg, W1 + (size_t)l * DD * DD2, b1 + (size_t)l * DD2, hm);

        // hn = hm @ W2[l] + b2[l]          [N,256] x [256,128]
        dim3 g2(N / 16, DD / 64);
        gemm_tdm_kernel<DD2, DD, false><<<g2, 128, 0, stream>>>(
            hm, W2 + (size_t)l * DD2 * DD, b2 + (size_t)l * DD, hn);

        // BatchNorm (train statistics)
        bn_stats_kernel<<<DD, 256, 0, stream>>>(hn, stats, N);

        float* dst = (l == NLAYER - 1) ? (float*)d_out : h;
        int tot = N * (DD / 4);
        bn_apply_kernel<<<(tot + 255) / 256, 256, 0, stream>>>(
            hn, stats, gamma + (size_t)l * DD, beta + (size_t)l * DD,
            dst, N, (l < NLAYER - 1) ? 1 : 0);
    }
}